// SmoothLoss_59158879535788
// MI455X (gfx1250) — compile-verified
//
#include <hip/hip_runtime.h>
#include <hip/hip_bf16.h>
#include <math.h>
#include <stdint.h>

// SmoothLoss on MI455X (gfx1250): memory-bound streaming reduction.
// ~224 MB of reads @ 23.3 TB/s => ~9.6 us floor. No matmul structure -> no
// WMMA; the CDNA5-specific path used here is the async data-mover:
// global_load_async_to_lds_b32 (ASYNCcnt) stages each 256-node weight tile
// into LDS (non-temporal: single-use stream, keep L2 for the hot labels
// array) while edge-index/label gathers run on LOADcnt in parallel.

#define BS     256
#define KEDGE  9
#define TILE_BYTES (BS * KEDGE * 4)   // 9216 B per tile

__global__ __launch_bounds__(BS) void node_loss_kernel(
    const int* __restrict__ edge_index,   // (2, N*K) int32
    const float* __restrict__ w,          // (N*K,)  f32
    const int* __restrict__ labels,       // (N,)    int32
    int N,
    float* __restrict__ partials)         // (2*gridDim.x,) f32
{
    const int E = N * KEDGE;              // 18M < 2^31
    const int* __restrict__ src = edge_index;
    const int* __restrict__ tgt = edge_index + E;

    __shared__ __align__(16) float wbuf[BS * KEDGE];   // 9216 B
    __shared__ float red_s[BS / 32];
    __shared__ float red_r[BS / 32];

    const unsigned tid = threadIdx.x;
    // Low 32 bits of a flat pointer to __shared__ == allocation-relative LDS
    // byte offset (flat-LDS aperture mapping, ISA 10.2) -> feed directly as
    // the async instruction's LDS-address VGPR.
    const uint32_t lds_addr = (uint32_t)(uintptr_t)(&wbuf[0]) + tid * 4u;

    float acc_s = 0.0f;   // per-thread smoothness partial
    float acc_r = 0.0f;   // per-thread rank-hinge partial

    const int full_tiles = N / BS;
    for (int tile = blockIdx.x; tile < full_tiles; tile += gridDim.x) {
        // ---- async copy: tile's 9216 B of weights, flat layout, coalesced
        // (dense 128 B per wave per op). INST_OFFSET is added to BOTH the
        // global and LDS address, so one (lds_addr, voff) pair covers all 9
        // chunks of 1024 B. Non-temporal: this stream is read exactly once.
        const uint32_t voff = (uint32_t)tile * (uint32_t)TILE_BYTES + tid * 4u;
        asm volatile(
            "global_load_async_to_lds_b32 %0, %1, %2 offset:0 th:TH_LOAD_NT\n\t"
            "global_load_async_to_lds_b32 %0, %1, %2 offset:1024 th:TH_LOAD_NT\n\t"
            "global_load_async_to_lds_b32 %0, %1, %2 offset:2048 th:TH_LOAD_NT\n\t"
            "global_load_async_to_lds_b32 %0, %1, %2 offset:3072 th:TH_LOAD_NT\n\t"
            "global_load_async_to_lds_b32 %0, %1, %2 offset:4096 th:TH_LOAD_NT\n\t"
            "global_load_async_to_lds_b32 %0, %1, %2 offset:5120 th:TH_LOAD_NT\n\t"
            "global_load_async_to_lds_b32 %0, %1, %2 offset:6144 th:TH_LOAD_NT\n\t"
            "global_load_async_to_lds_b32 %0, %1, %2 offset:7168 th:TH_LOAD_NT\n\t"
            "global_load_async_to_lds_b32 %0, %1, %2 offset:8192 th:TH_LOAD_NT"
            :
            : "v"(lds_addr), "v"(voff), "s"(w)
            : "memory");

        // ---- overlap: edge-index loads (streamed once -> non-temporal) and
        // label gathers (hot in L2) while the async copy is in flight.
        const int node = tile * BS + (int)tid;
        const int e0 = node * KEDGE;
        const int* __restrict__ sp = src + e0;
        const int* __restrict__ tp = tgt + e0;
        int pos_mask = 0;
#pragma unroll
        for (int k = 0; k < KEDGE; ++k) {
            const int si = __builtin_nontemporal_load(sp + k);
            const int ti = __builtin_nontemporal_load(tp + k);
            pos_mask |= (labels[si] == labels[ti]) ? (1 << k) : 0;
        }

        asm volatile("s_wait_asynccnt 0" ::: "memory");
        __syncthreads();   // tile data written by all waves' async ops

        // stride-9 word reads: gcd(9,64)==1 -> bank-conflict-free
        const float* myw = &wbuf[tid * KEDGE];
        float sum_p = 0.0f, sum_n = 0.0f;
        float mx = -__builtin_inff(), mn = __builtin_inff();
#pragma unroll
        for (int k = 0; k < KEDGE; ++k) {
            const float wv = myw[k];
            if (pos_mask & (1 << k)) { sum_p += wv; mx = fmaxf(mx, wv); }
            else                     { sum_n += wv; mn = fminf(mn, wv); }
        }
        __syncthreads();   // all reads done before next tile's async writes

        const int pc = __popc(pos_mask);
        acc_s += expf(sum_p / (float)pc) + expf(-sum_n / (float)(KEDGE - pc));
        acc_r += fmaxf(mx - mn, 0.0f);
    }

    // ---- tail (N % BS nodes): block 0, direct global loads (no async OOB)
    if (blockIdx.x == 0) {
        const int node = full_tiles * BS + (int)tid;
        if (node < N) {
            const int e0 = node * KEDGE;
            float sum_p = 0.0f, sum_n = 0.0f;
            float mx = -__builtin_inff(), mn = __builtin_inff();
            int pc = 0;
#pragma unroll
            for (int k = 0; k < KEDGE; ++k) {
                const float wv = w[e0 + k];
                const bool p = (labels[src[e0 + k]] == labels[tgt[e0 + k]]);
                if (p) { ++pc; sum_p += wv; mx = fmaxf(mx, wv); }
                else   { sum_n += wv; mn = fminf(mn, wv); }
            }
            acc_s += expf(sum_p / (float)pc) +
                     expf(-sum_n / (float)(KEDGE - pc));
            acc_r += fmaxf(mx - mn, 0.0f);
        }
    }

    // ---- wave32 reduction, then cross-wave via LDS
#pragma unroll
    for (int off = 16; off > 0; off >>= 1) {
        acc_s += __shfl_xor(acc_s, off, 32);
        acc_r += __shfl_xor(acc_r, off, 32);
    }
    if ((tid & 31u) == 0u) {
        red_s[tid >> 5] = acc_s;
        red_r[tid >> 5] = acc_r;
    }
    __syncthreads();
    if (tid == 0) {
        float bsum = 0.0f, rsum = 0.0f;
#pragma unroll
        for (int i = 0; i < BS / 32; ++i) { bsum += red_s[i]; rsum += red_r[i]; }
        partials[2 * blockIdx.x]     = bsum;
        partials[2 * blockIdx.x + 1] = rsum;
    }
}

__global__ __launch_bounds__(256) void finalize_kernel(
    const float* __restrict__ partials, int nparts, int N,
    float* __restrict__ out)
{
    __shared__ double sh_s[256];
    __shared__ double sh_r[256];
    double s = 0.0, r = 0.0;
    for (int i = (int)threadIdx.x; i < nparts; i += 256) {
        s += (double)partials[2 * i];
        r += (double)partials[2 * i + 1];
    }
    sh_s[threadIdx.x] = s;
    sh_r[threadIdx.x] = r;
    __syncthreads();
    for (int off = 128; off > 0; off >>= 1) {
        if ((int)threadIdx.x < off) {
            sh_s[threadIdx.x] += sh_s[threadIdx.x + off];
            sh_r[threadIdx.x] += sh_r[threadIdx.x + off];
        }
        __syncthreads();
    }
    if (threadIdx.x == 0) {
        const double S = sh_s[0];
        const double R = sh_r[0] / (double)N;
        out[0] = (float)(log(S + 2.0) + 2.0 * exp(R));
    }
}

extern "C" void kernel_launch(void* const* d_in, const int* in_sizes, int n_in,
                              void* d_out, int out_size, void* d_ws, size_t ws_size,
                              hipStream_t stream) {
    (void)n_in; (void)out_size; (void)ws_size;
    const int*   edge_index = (const int*)d_in[0];    // (2, N*K) int32
    const float* edge_w     = (const float*)d_in[1];  // (1, N*K) f32
    const int*   labels     = (const int*)d_in[2];    // (N,)     int32
    const int N = in_sizes[2];

    float* partials = (float*)d_ws;                   // 2*grid floats

    int full_tiles = N / BS;
    int grid = full_tiles < 1 ? 1 : (full_tiles > 1024 ? 1024 : full_tiles);

    node_loss_kernel<<<grid, BS, 0, stream>>>(edge_index, edge_w, labels, N,
                                              partials);
    finalize_kernel<<<1, 256, 0, stream>>>(partials, grid, N, (float*)d_out);
}